// LosslessLatentDecoder_84645215469848
// MI455X (gfx1250) — compile-verified
//
#include <hip/hip_runtime.h>

// LosslessLatentDecoder: conv_transpose2d with a permutation kernel ==
// depth-to-space pixel shuffle:
//   out[b, c, 4h+y, 4w+x] = in[b, c*16 + 4x + y, h, w]
// B=8, Cin=64, H=W=256, d=4, Cout=4, H'=W'=1024.  Pure data movement:
// 256 MiB traffic, 0 FLOPs -> HBM-bound (~11.5 us at 23.3 TB/s).
// Strategy: b128 loads/stores, fully coalesced, non-temporal hints
// (footprint > 192MB L2, zero reuse).

typedef __attribute__((ext_vector_type(4))) float v4f;

__global__ __launch_bounds__(256) void
LosslessLatentDecoder_kernel(const float* __restrict__ in, float* __restrict__ out) {
    const int tid = blockIdx.x * 256 + threadIdx.x;

    // Decompose tid -> (b, c, y, h, t) with t = w/4 (w-group of 4).
    // Consecutive tids -> consecutive t -> coalesced VMEM.
    const int t = tid & 63;           // w = 4t .. 4t+3   (W/4 = 64)
    const int h = (tid >> 6) & 255;   // H = 256
    const int y = (tid >> 14) & 3;    // d = 4
    const int c = (tid >> 16) & 3;    // Cout = 4
    const int b = tid >> 18;          // B = 8

    const v4f* __restrict__ inv  = (const v4f*)in;
    v4f*       __restrict__ outv = (v4f*)out;

    // Input plane for x: p = b*64 + c*16 + 4x + y. Row of W=256 floats = 64 v4f.
    const int p0   = (b << 6) + (c << 4) + y;       // x = 0 plane
    const int in0  = (p0 * 256 + h) * 64 + t;       // v4f index
    const int pstr = 4 * 256 * 64;                  // +4 planes per x-step (v4f units)

    // 4 coalesced b128 loads, one per x-plane. NT: no reuse anywhere.
    const v4f v0 = __builtin_nontemporal_load(inv + in0);
    const v4f v1 = __builtin_nontemporal_load(inv + in0 + 1 * pstr);
    const v4f v2 = __builtin_nontemporal_load(inv + in0 + 2 * pstr);
    const v4f v3 = __builtin_nontemporal_load(inv + in0 + 3 * pstr);

    // Output: row (b*4+c, oh=4h+y) of 1024 floats = 256 v4f; this thread
    // owns ow = 16t .. 16t+15  -> 4 consecutive v4f stores (64B contiguous).
    const int o0 = ((((b << 2) + c) << 10) + ((h << 2) + y)) * 256 + (t << 2);

    // 4x4 register transpose: output v4f j = { v0[j], v1[j], v2[j], v3[j] }
    v4f o;
    o = (v4f){v0.x, v1.x, v2.x, v3.x};
    __builtin_nontemporal_store(o, outv + o0 + 0);
    o = (v4f){v0.y, v1.y, v2.y, v3.y};
    __builtin_nontemporal_store(o, outv + o0 + 1);
    o = (v4f){v0.z, v1.z, v2.z, v3.z};
    __builtin_nontemporal_store(o, outv + o0 + 2);
    o = (v4f){v0.w, v1.w, v2.w, v3.w};
    __builtin_nontemporal_store(o, outv + o0 + 3);
}

extern "C" void kernel_launch(void* const* d_in, const int* in_sizes, int n_in,
                              void* d_out, int out_size, void* d_ws, size_t ws_size,
                              hipStream_t stream) {
    const float* x   = (const float*)d_in[0];   // [8, 64, 256, 256] f32
    // d_in[1] (permutation kernel) and d_in[2] (latent_depth=4) are fixed by
    // construction; the permutation is hard-wired into the index math above.
    float* out = (float*)d_out;                 // [8, 4, 1024, 1024] f32

    const int total_threads = 8 * 4 * 4 * 256 * 64;  // 2,097,152
    LosslessLatentDecoder_kernel<<<total_threads / 256, 256, 0, stream>>>(x, out);
}